// TtMambaBlock_17514876634004
// MI455X (gfx1250) — compile-verified
//
#include <hip/hip_runtime.h>
#include <math.h>

typedef __attribute__((ext_vector_type(2))) float v2f;
typedef __attribute__((ext_vector_type(8))) float v8f;
typedef __attribute__((ext_vector_type(4))) int   v4i;

#define BATCH   32
#define DMODEL  2560
#define DINNER  5120
#define NSTATE  16
#define DTRANK  160
#define XDB     (DTRANK + 2*NSTATE)   // 192

// ---------------------------------------------------------------------------
// CDNA5 async global->LDS copy (ASYNCcnt-tracked), 16 bytes per lane.
// Builtin signature (from clang diagnostic): (as1 v4i*, as3 v4i*, imm, imm).
// ---------------------------------------------------------------------------
__device__ __forceinline__ void async_copy16(const float* gp, float* lp)
{
#if __has_builtin(__builtin_amdgcn_global_load_async_to_lds_b128)
    __builtin_amdgcn_global_load_async_to_lds_b128(
        (__attribute__((address_space(1))) v4i*)gp,
        (__attribute__((address_space(3))) v4i*)lp,
        0, 0);
#else
    asm volatile("global_load_async_to_lds_b128 %0, %1, off"
                 :: "v"((unsigned)(size_t)lp),
                    "v"((unsigned long long)(size_t)gp)
                 : "memory");
#endif
}

template<int Nw>
__device__ __forceinline__ void wait_async()
{
#if __has_builtin(__builtin_amdgcn_s_wait_asynccnt)
    __builtin_amdgcn_s_wait_asynccnt((unsigned short)Nw);
#else
    asm volatile("s_wait_asynccnt %0" :: "i"(Nw) : "memory");
#endif
}

// ---------------------------------------------------------------------------
// Skinny fp32 WMMA GEMM slice:  Cslice[32][NN] = A[32][k0:k0+KS] * W[:, k0:k0+KS]^T
// KS = K / gridDim.y (contiguous split-K slice per blockIdx.y).
// Block = 256 threads (8 wave32); each wave: one 16-col group, two 16-row M tiles.
// Triple-buffered LDS fed by async global->LDS loads (5 per thread per chunk).
// ---------------------------------------------------------------------------
template<int KS, int NN, int LDA, int LDW>
__global__ __launch_bounds__(256) void gemm32_async(
    const float* __restrict__ A,
    const float* __restrict__ W,
    float* __restrict__ C)
{
    constexpr int KC   = 32;
    constexpr int NCH  = KS / KC;
    constexpr int NBUF = 3;
    static_assert(KS % KC == 0, "KS must be a multiple of 32");

    __shared__ __align__(16) float As[NBUF][32][KC + 4];
    __shared__ __align__(16) float Ws[NBUF][128][KC + 4];

    const int tid   = threadIdx.x;
    const int wave  = tid >> 5;
    const int lane  = tid & 31;
    const int lr    = lane & 15;
    const int half  = lane >> 4;
    const int nblk  = blockIdx.x * 128;
    const int ncol0 = nblk + wave * 16;

    A += (size_t)blockIdx.y * KS;                 // K-slice
    W += (size_t)blockIdx.y * KS;
    C += (size_t)blockIdx.y * 32 * NN;            // partial slab

    const int arow = tid >> 3;                    // A staging: 32 rows x 8 float4
    const int ac4  = tid & 7;

    auto issue = [&](int chunk, int buf) {
        const int k0 = chunk * KC;
        async_copy16(A + (size_t)arow * LDA + k0 + ac4 * 4, &As[buf][arow][ac4 * 4]);
        #pragma unroll
        for (int i = 0; i < 4; ++i) {             // W staging: 128 rows x 8 float4
            const int id  = tid + i * 256;
            const int row = id >> 3;
            const int c4  = id & 7;
            int gn = nblk + row;                  // branchless clamp (dup loads OOB)
            gn = gn < NN ? gn : NN - 1;
            async_copy16(W + (size_t)gn * LDW + k0 + c4 * 4, &Ws[buf][row][c4 * 4]);
        }
    };

    issue(0, 0);
    if (NCH > 1) issue(1, 1);
    if (NCH > 2) issue(2, 2);

    v8f c0 = {};   // M rows 0..15
    v8f c1 = {};   // M rows 16..31

    int buf = 0;
    for (int i = 0; i < NCH; ++i) {
        if      (i + 2 < NCH) wait_async<10>();   // 2 chunks still in flight
        else if (i + 1 < NCH) wait_async<5>();    // 1 chunk in flight
        else                  wait_async<0>();
        __syncthreads();

        #pragma unroll
        for (int kk = 0; kk < KC; kk += 4) {
            const int ka = kk + 2 * half;         // lanes 16..31 take K+2,K+3
            v2f a0 = { As[buf][lr][ka],      As[buf][lr][ka + 1]      };
            v2f a1 = { As[buf][lr + 16][ka], As[buf][lr + 16][ka + 1] };
            v2f bf = { Ws[buf][(wave << 4) + lr][ka], Ws[buf][(wave << 4) + lr][ka + 1] };
            c0 = __builtin_amdgcn_wmma_f32_16x16x4_f32(false, a0, false, bf, (short)0, c0, false, false);
            c1 = __builtin_amdgcn_wmma_f32_16x16x4_f32(false, a1, false, bf, (short)0, c1, false, false);
        }
        __syncthreads();                          // everyone done reading buf

        if (i + 3 < NCH) issue(i + 3, buf);       // refill this buffer
        buf = (buf + 1 == NBUF) ? 0 : buf + 1;
    }

    if (ncol0 < NN) {
        #pragma unroll
        for (int r = 0; r < 8; ++r) {
            const int m = r + 8 * half;           // C/D layout: VGPR r -> M=r (+8 upper half)
            C[(size_t)m        * NN + ncol0 + lr] = c0[r];
            C[(size_t)(m + 16) * NN + ncol0 + lr] = c1[r];
        }
    }
}

// ---------------------------------------------------------------------------
// Sum SPLIT partial slabs (L2-resident) into the final matrix.
// ---------------------------------------------------------------------------
template<int SPLIT>
__global__ __launch_bounds__(256) void reduce_kernel(
    const float* __restrict__ part, float* __restrict__ out, int n)
{
    const int idx = blockIdx.x * blockDim.x + threadIdx.x;
    if (idx >= n) return;
    float a = 0.0f;
    #pragma unroll
    for (int s = 0; s < SPLIT; ++s) a += part[(size_t)s * n + idx];
    out[idx] = a;
}

// ---------------------------------------------------------------------------
// conv shift-register (k=4) + bias + SiLU
// ---------------------------------------------------------------------------
__global__ __launch_bounds__(256) void conv_kernel(
    const float* __restrict__ cs,   // (4, B, DINNER)
    const float* __restrict__ cw,   // (4, DINNER)
    const float* __restrict__ cb,   // (DINNER)
    const float* __restrict__ xs,   // (B, DINNER)
    float* __restrict__ conv)       // (B, DINNER)
{
    const int idx = blockIdx.x * blockDim.x + threadIdx.x;
    if (idx >= BATCH * DINNER) return;
    const int d = idx % DINNER;
    const int S = BATCH * DINNER;
    float v = cs[S     + idx] * cw[d]
            + cs[2 * S + idx] * cw[DINNER + d]
            + cs[3 * S + idx] * cw[2 * DINNER + d]
            + xs[idx]         * cw[3 * DINNER + d]
            + cb[d];
    conv[idx] = v / (1.0f + __expf(-v));   // silu
}

// ---------------------------------------------------------------------------
// SSM decode step + silu gate
// ---------------------------------------------------------------------------
__global__ __launch_bounds__(256) void ssm_kernel(
    const float* __restrict__ dt_raw,   // (B, DINNER) pre-bias
    const float* __restrict__ dt_bias,  // (DINNER)
    const float* __restrict__ conv,     // (B, DINNER)
    const float* __restrict__ x_db,     // (B, 192)
    const float* __restrict__ A_log,    // (DINNER, N)
    const float* __restrict__ Dv,       // (DINNER)
    const float* __restrict__ st,       // (B, DINNER, N)
    const float* __restrict__ res,      // (B, DINNER)
    float* __restrict__ g)              // (B, DINNER)
{
    const int idx = blockIdx.x * blockDim.x + threadIdx.x;
    if (idx >= BATCH * DINNER) return;
    const int b = idx / DINNER;
    const int d = idx % DINNER;

    float dtv = dt_raw[idx] + dt_bias[d];
    dtv = (dtv > 20.0f) ? dtv : logf(1.0f + expf(dtv));   // softplus

    const float cv = conv[idx];
    const float* __restrict__ Bm = x_db + (size_t)b * XDB + DTRANK;
    const float* __restrict__ Cm = Bm + NSTATE;
    const float* __restrict__ sp = st + (size_t)idx * NSTATE;
    const float* __restrict__ Al = A_log + (size_t)d * NSTATE;

    float acc = 0.0f;
    #pragma unroll
    for (int n = 0; n < NSTATE; ++n) {
        const float dA = __expf(dtv * (-__expf(Al[n])));
        const float h  = sp[n] * dA + dtv * Bm[n] * cv;
        acc += h * Cm[n];
    }
    const float y = acc + Dv[d] * cv;
    const float r = res[idx];
    g[idx] = y * (r / (1.0f + __expf(-r)));
}

// ---------------------------------------------------------------------------
extern "C" void kernel_launch(void* const* d_in, const int* in_sizes, int n_in,
                              void* d_out, int out_size, void* d_ws, size_t ws_size,
                              hipStream_t stream)
{
    (void)in_sizes; (void)n_in; (void)out_size; (void)ws_size;

    const float* x           = (const float*)d_in[0];
    const float* conv_states = (const float*)d_in[1];
    const float* conv_w      = (const float*)d_in[2];
    const float* conv_b      = (const float*)d_in[3];
    const float* W_ssm_in    = (const float*)d_in[4];
    const float* W_mlp       = (const float*)d_in[5];
    const float* W_out       = (const float*)d_in[6];
    const float* W_x_proj    = (const float*)d_in[7];
    const float* W_dt        = (const float*)d_in[8];
    const float* dt_bias     = (const float*)d_in[9];
    const float* A_log       = (const float*)d_in[10];
    const float* Dv          = (const float*)d_in[11];
    const float* ssm_state   = (const float*)d_in[12];
    float* out               = (float*)d_out;

    float* ws     = (float*)d_ws;
    float* xs     = ws;                              // 32*5120
    float* res    = xs     + BATCH * DINNER;
    float* conv   = res    + BATCH * DINNER;
    float* x_db   = conv   + BATCH * DINNER;         // 32*192
    float* dt_raw = x_db   + BATCH * XDB;
    float* g      = dt_raw + BATCH * DINNER;
    float* part   = g      + BATCH * DINNER;         // 8*32*5120 partial slabs

    const int EW = (BATCH * DINNER + 255) / 256;

    // 1) xs = x @ W_ssm_in^T   (K=2560, N=5120, split-K 8)
    gemm32_async<320, DINNER, DMODEL, DMODEL>
        <<<dim3(DINNER / 128, 8), 256, 0, stream>>>(x, W_ssm_in, part);
    reduce_kernel<8><<<EW, 256, 0, stream>>>(part, xs, BATCH * DINNER);

    // 2) res = x @ W_mlp^T
    gemm32_async<320, DINNER, DMODEL, DMODEL>
        <<<dim3(DINNER / 128, 8), 256, 0, stream>>>(x, W_mlp, part);
    reduce_kernel<8><<<EW, 256, 0, stream>>>(part, res, BATCH * DINNER);

    // 3) conv = silu(shift-conv(xs) + bias)
    conv_kernel<<<EW, 256, 0, stream>>>(conv_states, conv_w, conv_b, xs, conv);

    // 4) x_db = conv @ W_x_proj^T   (K=5120, N=192, split-K 8)
    gemm32_async<640, XDB, DINNER, DINNER>
        <<<dim3(2, 8), 256, 0, stream>>>(conv, W_x_proj, part);
    reduce_kernel<8><<<(BATCH * XDB + 255) / 256, 256, 0, stream>>>(part, x_db, BATCH * XDB);

    // 5) dt_raw = x_db[:, :160] @ W_dt^T   (K=160, N=5120, split-K 5)
    gemm32_async<32, DINNER, XDB, DTRANK>
        <<<dim3(DINNER / 128, 5), 256, 0, stream>>>(x_db, W_dt, part);
    reduce_kernel<5><<<EW, 256, 0, stream>>>(part, dt_raw, BATCH * DINNER);

    // 6) SSM step + gate
    ssm_kernel<<<EW, 256, 0, stream>>>(dt_raw, dt_bias, conv, x_db, A_log, Dv,
                                       ssm_state, res, g);

    // 7) out = g @ W_out^T   (K=5120, N=2560, split-K 8)
    gemm32_async<640, DMODEL, DINNER, DINNER>
        <<<dim3(DMODEL / 128, 8), 256, 0, stream>>>(g, W_out, part);
    reduce_kernel<8><<<(BATCH * DMODEL + 255) / 256, 256, 0, stream>>>(part, out, BATCH * DMODEL);
}